// Invariant_Capture_55018531062591
// MI455X (gfx1250) — compile-verified
//
#include <hip/hip_runtime.h>
#include <hip/hip_bf16.h>
#include <math.h>

#define N_ENT  40000
#define N_EDGE 640000
#define CH     128
#define N_REL  50
#define NPAD   64   // N_REL padded to 4 WMMA tiles of 16

typedef float v2f __attribute__((ext_vector_type(2)));
typedef float v8f __attribute__((ext_vector_type(8)));

// order-preserving float <-> uint bijection for atomicMax-based segment max
__device__ __forceinline__ unsigned f2ord(float f) {
  unsigned u = __float_as_uint(f);
  return (u & 0x80000000u) ? ~u : (u | 0x80000000u);
}
__device__ __forceinline__ float ord2f(unsigned u) {
  return (u & 0x80000000u) ? __uint_as_float(u & 0x7FFFFFFFu) : __uint_as_float(~u);
}

// ---------------- init: max = -inf (ordered), sums = 0 ----------------
__global__ void k_init(unsigned* max1, float* sum1, unsigned* max2, float* sum2) {
  int i = blockIdx.x * blockDim.x + threadIdx.x;
  if (i < N_ENT) {
    max1[i] = 0x007FFFFFu;  // f2ord(-inf)
    max2[i] = 0x007FFFFFu;
    sum1[i] = 0.0f;
    sum2[i] = 0.0f;
  }
}

// ---------------- K50 = R @ W  (50 x 128) ----------------
__global__ void k_rel_proj(const float* __restrict__ R, const float* __restrict__ W,
                           float* __restrict__ K50) {
  int o = blockIdx.x * blockDim.x + threadIdx.x;
  if (o >= N_REL * CH) return;
  int n = o >> 7, k = o & 127;
  float acc = 0.0f;
  for (int j = 0; j < CH; ++j) acc = fmaf(R[n * CH + j], W[j * CH + k], acc);
  K50[o] = acc;
}

// ---------------- rn2[n] = ||R_n||^2 ----------------
__global__ void k_rnorm(const float* __restrict__ R, float* __restrict__ rn2) {
  int n = threadIdx.x;
  if (n >= NPAD) return;
  float acc = 0.0f;
  if (n < N_REL)
    for (int j = 0; j < CH; ++j) { float v = R[n * CH + j]; acc = fmaf(v, v, acc); }
  rn2[n] = acc;
}

// ---------------- M2p[k][n] = (1/16) * sum_j W[k][j] * K50[n][j], padded to 64 cols ----------------
__global__ void k_m2(const float* __restrict__ W, const float* __restrict__ K50,
                     float* __restrict__ M2p) {
  int o = blockIdx.x * blockDim.x + threadIdx.x;
  if (o >= CH * NPAD) return;
  int k = o >> 6, n = o & 63;
  float acc = 0.0f;
  if (n < N_REL) {
    for (int j = 0; j < CH; ++j) acc = fmaf(W[k * CH + j], K50[n * CH + j], acc);
    acc *= 0.0625f;  // / (sqrt(64) * 2 heads)
  }
  M2p[o] = acc;
}

// ---------------- S = E @ M2p  via V_WMMA_F32_16X16X4_F32 ----------------
// 128 threads = 4 waves; each wave: one 16-row M tile x four 16-col N tiles.
// B is staged in LDS pair-interleaved so every fragment is one ds_load_b64.
__global__ __launch_bounds__(128) void k_gemm(const float* __restrict__ E,
                                              const float* __restrict__ M2p,
                                              float* __restrict__ S) {
  __shared__ float sA[64 * CH];            // 32 KB: 64 entity rows, row-major
  __shared__ v2f   sB2[(CH / 2) * NPAD];   // 32 KB: sB2[p*NPAD+n] = {B[2p][n], B[2p+1][n]}
  int tid = threadIdx.x;
  int rowblk = blockIdx.x * 64;

  // stage A (coalesced float4)
  const float4* Eg = (const float4*)(E + (size_t)rowblk * CH);
  float4* sA4 = (float4*)sA;
#pragma unroll
  for (int i = 0; i < (64 * CH / 4) / 128; ++i) sA4[tid + i * 128] = Eg[tid + i * 128];

  // stage B pair-interleaved: pair p covers rows 2p, 2p+1
#pragma unroll
  for (int it = 0; it < (CH / 2) * NPAD / 128; ++it) {
    int i = tid + it * 128;
    int p = i >> 6, n = i & 63;
    v2f v;
    v.x = M2p[(2 * p) * NPAD + n];
    v.y = M2p[(2 * p + 1) * NPAD + n];
    sB2[i] = v;
  }
  __syncthreads();

  int wave = tid >> 5, lane = tid & 31;
  int lr = lane & 15, hi = lane >> 4;  // hi: lanes 16-31 carry K+2 / M+8 halves
  const float* Arow = &sA[(wave * 16 + lr) * CH];

  v8f acc0 = {}, acc1 = {}, acc2 = {}, acc3 = {};
#pragma unroll 4
  for (int k0 = 0; k0 < CH; k0 += 4) {
    // A 16x4: lane(row=lr) holds K = k0+2*hi, +1  -> one b64 load
    v2f a = *(const v2f*)(Arow + k0 + (hi << 1));
    // B 4x16 per N-tile: lanes 0-15 rows (k0,k0+1) = pair k0/2; lanes 16-31 pair k0/2+1
    const v2f* bp = &sB2[((k0 >> 1) + hi) * NPAD];
    v2f b0 = bp[lr];
    v2f b1 = bp[16 + lr];
    v2f b2 = bp[32 + lr];
    v2f b3 = bp[48 + lr];
    acc0 = __builtin_amdgcn_wmma_f32_16x16x4_f32(false, a, false, b0, (short)0, acc0, false, false);
    acc1 = __builtin_amdgcn_wmma_f32_16x16x4_f32(false, a, false, b1, (short)0, acc1, false, false);
    acc2 = __builtin_amdgcn_wmma_f32_16x16x4_f32(false, a, false, b2, (short)0, acc2, false, false);
    acc3 = __builtin_amdgcn_wmma_f32_16x16x4_f32(false, a, false, b3, (short)0, acc3, false, false);
  }

  // C/D layout: VGPR v, lanes 0-15 -> M=v, lanes 16-31 -> M=v+8; N = tile*16 + lr
  int gr = rowblk + wave * 16;
#pragma unroll
  for (int v = 0; v < 8; ++v) {
    int m = v + 8 * hi;
    float* out = &S[(size_t)(gr + m) * NPAD + lr];
    out[0]  = acc0[v];
    out[16] = acc1[v];
    out[32] = acc2[v];
    out[48] = acc3[v];
  }
}

// ---------------- softmax #1: segment max then exp-sum over head ----------------
__global__ void k_max1(const int* __restrict__ head, const int* __restrict__ etype,
                       const float* __restrict__ S, unsigned* __restrict__ max1) {
  int e = blockIdx.x * blockDim.x + threadIdx.x;
  if (e >= N_EDGE) return;
  int h = head[e];
  float sr = S[(size_t)h * NPAD + (etype[e] - 1)];
  atomicMax(&max1[h], f2ord(sr));
}

__global__ void k_sum1(const int* __restrict__ head, const int* __restrict__ etype,
                       const float* __restrict__ S, const unsigned* __restrict__ max1,
                       float* __restrict__ sum1) {
  int e = blockIdx.x * blockDim.x + threadIdx.x;
  if (e >= N_EDGE) return;
  int h = head[e];
  float sr = S[(size_t)h * NPAD + (etype[e] - 1)];
  atomicAdd(&sum1[h], __expf(sr - ord2f(max1[h])));
}

// ---------------- score_trip = h.t + rel_score^2 * ||r||^2 ; start softmax #2 max ----------------
__global__ __launch_bounds__(256) void k_trip(const int* __restrict__ head,
                                              const int* __restrict__ tail,
                                              const int* __restrict__ etype,
                                              const float* __restrict__ E,
                                              const float* __restrict__ S,
                                              const unsigned* __restrict__ max1,
                                              const float* __restrict__ sum1,
                                              const float* __restrict__ rn2,
                                              float* __restrict__ trip,
                                              unsigned* __restrict__ max2) {
  int lane = threadIdx.x & 31;
  int e = blockIdx.x * 8 + (threadIdx.x >> 5);  // one wave32 per edge
  if (e >= N_EDGE) return;
  int h = head[e], t = tail[e];
  const float4* hp = (const float4*)(E + (size_t)h * CH);
  const float4* tp = (const float4*)(E + (size_t)t * CH);
  float4 hv = hp[lane], tv = tp[lane];
  float d = hv.x * tv.x + hv.y * tv.y + hv.z * tv.z + hv.w * tv.w;
#pragma unroll
  for (int off = 16; off; off >>= 1) d += __shfl_xor(d, off, 32);
  if (lane == 0) {
    int rt = etype[e] - 1;
    float sr = S[(size_t)h * NPAD + rt];
    float rs = __expf(sr - ord2f(max1[h])) / sum1[h];
    float sc = d + rs * rs * rn2[rt];
    trip[e] = sc;
    atomicMax(&max2[h], f2ord(sc));
  }
}

// ---------------- softmax #2: exp-sum, write numerator into d_out ----------------
__global__ void k_sum2(const int* __restrict__ head, const float* __restrict__ trip,
                       const unsigned* __restrict__ max2, float* __restrict__ sum2,
                       float* __restrict__ out) {
  int e = blockIdx.x * blockDim.x + threadIdx.x;
  if (e >= N_EDGE) return;
  int h = head[e];
  float e2 = __expf(trip[e] - ord2f(max2[h]));
  out[e] = e2;
  atomicAdd(&sum2[h], e2);
}

__global__ void k_norm(const int* __restrict__ head, const float* __restrict__ sum2,
                       float* __restrict__ out) {
  int e = blockIdx.x * blockDim.x + threadIdx.x;
  if (e >= N_EDGE) return;
  out[e] /= sum2[head[e]];
}

extern "C" void kernel_launch(void* const* d_in, const int* in_sizes, int n_in,
                              void* d_out, int out_size, void* d_ws, size_t ws_size,
                              hipStream_t stream) {
  const float* E  = (const float*)d_in[0];   // entity_emb [40000,128]
  // d_in[1] = user_emb (unused by reference output)
  const float* R  = (const float*)d_in[2];   // relation_weight [50,128]
  const float* W  = (const float*)d_in[3];   // kg_W_r [128,128]
  const int* eidx  = (const int*)d_in[4];    // edge_index [2, 640000]
  const int* etype = (const int*)d_in[5];    // edge_type [640000]
  const int* head = eidx;
  const int* tail = eidx + N_EDGE;
  float* out = (float*)d_out;

  // workspace layout (float units)
  float* ws   = (float*)d_ws;
  float* K50  = ws;                                   // 50*128 = 6400
  float* rn2  = ws + 6400;                            // 64
  float* M2p  = ws + 6464;                            // 128*64 = 8192
  float* S    = ws + 14656;                           // 40000*64 = 2,560,000
  unsigned* max1 = (unsigned*)(ws + 14656 + (size_t)N_ENT * NPAD);
  float*    sum1 = (float*)max1 + N_ENT;
  unsigned* max2 = (unsigned*)((float*)max1 + 2 * N_ENT);
  float*    sum2 = (float*)max1 + 3 * N_ENT;
  float*    trip = (float*)max1 + 4 * N_ENT;          // 640,000 -> total ~13.5 MB

  k_init<<<(N_ENT + 255) / 256, 256, 0, stream>>>(max1, sum1, max2, sum2);
  k_rel_proj<<<(N_REL * CH + 255) / 256, 256, 0, stream>>>(R, W, K50);
  k_rnorm<<<1, 64, 0, stream>>>(R, rn2);
  k_m2<<<(CH * NPAD + 255) / 256, 256, 0, stream>>>(W, K50, M2p);
  k_gemm<<<N_ENT / 64, 128, 0, stream>>>(E, M2p, S);
  k_max1<<<(N_EDGE + 255) / 256, 256, 0, stream>>>(head, etype, S, max1);
  k_sum1<<<(N_EDGE + 255) / 256, 256, 0, stream>>>(head, etype, S, max1, sum1);
  k_trip<<<N_EDGE / 8, 256, 0, stream>>>(head, tail, etype, E, S, max1, sum1, rn2, trip, max2);
  k_sum2<<<(N_EDGE + 255) / 256, 256, 0, stream>>>(head, trip, max2, sum2, out);
  k_norm<<<(N_EDGE + 255) / 256, 256, 0, stream>>>(head, sum2, out);
}